// ModelNew_66924180406647
// MI455X (gfx1250) — compile-verified
//
#include <hip/hip_runtime.h>
#include <hip/hip_bf16.h>

// Fused causal relu-attention for B=8, T=1024, C=768, 12 heads, hs=64.
// Stage 1: qkv = x @ W_attn + b_attn  -> scattered into d_ws as
//          [3 (q/k/v)][B*nh = 96][T = 1024][hs = 64]   (75.5 MB scratch)
// Stage 2: per (head, q-tile-pair): S = Q Kt, scale+causal+relu, Y += S V
// All matrix math on the native fp32 WMMA pipe: V_WMMA_F32_16X16X4_F32.
// Working set (132MB) fits MI455X's 192MB L2; per-wave tiles are sized for
// low vmem-issue per WMMA, and __launch_bounds__(256, 2) raises the VGPR
// budget (~512) so the 16 fp32 accumulators do NOT spill to scratch.

typedef float v2f __attribute__((ext_vector_type(2)));
typedef float v8f __attribute__((ext_vector_type(8)));

#define BT     8192         // B*T
#define CIN    768
#define C3     2304
#define NH     12
#define HS     64
#define TT     1024
#define SCALE_ 0.125f       // 1/sqrt(64)

static __device__ __forceinline__ v8f wmma_f32(v2f a, v2f b, v8f c) {
    // 8 args: (neg_a, A, neg_b, B, c_mod, C, reuse_a, reuse_b)
    return __builtin_amdgcn_wmma_f32_16x16x4_f32(false, a, false, b, (short)0, c,
                                                 false, false);
}

// ---------------------------------------------------------------------------
// Kernel 1: QKV projection GEMM + bias, scatter to [3][96][1024][64].
// grid(16, 36), block 256 (8 waves). Wave tile = 64 rows x 64 cols
// (4x4 v8f accumulators = 128 VGPRs; needs the raised VGPR budget).
// Per K-step (K+=4): 4 x b64 (A) + 8 x b32 (B) feed 16 WMMAs.
// ---------------------------------------------------------------------------
__global__ __launch_bounds__(256, 2)
void attn_qkv_gemm_f32wmma(const float* __restrict__ x,
                           const float* __restrict__ W,
                           const float* __restrict__ bias,
                           float* __restrict__ qkv) {
    const int lane = threadIdx.x & 31;
    const int wave = threadIdx.x >> 5;
    const int hl   = lane >> 4;      // 0: lanes 0-15, 1: lanes 16-31
    const int l15  = lane & 15;

    const int m0 = blockIdx.x * 512 + wave * 64;   // output row base (BT dim)
    const int n0 = blockIdx.y * 64;                // output col base (3C dim)

    v8f acc[4][4] = {};
    // Single base pointers; per-tile displacements fold into imm offsets.
    const float* xrow = x + (size_t)(m0 + l15) * CIN;   // A rows, +mt*16*CIN imm
    const float* wp   = W + n0 + l15;                   // B cols, +nt*16 imm

    for (int k = 0; k < CIN; k += 4) {
        v2f a[4];
#pragma unroll
        for (int mt = 0; mt < 4; ++mt)
            a[mt] = *(const v2f*)(xrow + (size_t)mt * 16 * CIN + k + 2 * hl);
        __builtin_prefetch(xrow + k + 16, 0, 0);        // global_prefetch_b8
        const size_t wkk = (size_t)(k + 2 * hl) * C3;
#pragma unroll
        for (int nt = 0; nt < 4; ++nt) {
            v2f b;
            b.x = wp[wkk + nt * 16];
            b.y = wp[wkk + C3 + nt * 16];
#pragma unroll
            for (int mt = 0; mt < 4; ++mt)
                acc[mt][nt] = wmma_f32(a[mt], b, acc[mt][nt]);
        }
    }

    // Epilogue: bias + scatter into q/k/v [96][1024][64]
#pragma unroll
    for (int nt = 0; nt < 4; ++nt) {
        const int c3    = n0 + nt * 16 + l15;
        const float bv  = bias[c3];
        const int which = c3 / CIN;          // 0=q, 1=k, 2=v
        const int c     = c3 - which * CIN;
        const int h     = c >> 6;
        const int d     = c & 63;
        float* basep = qkv + (size_t)which * BT * CIN;
#pragma unroll
        for (int mt = 0; mt < 4; ++mt) {
#pragma unroll
            for (int r = 0; r < 8; ++r) {
                const int m  = m0 + mt * 16 + r + 8 * hl;  // row in BT
                const int b_ = m >> 10;
                const int t  = m & 1023;
                basep[((size_t)(b_ * NH + h) * TT + t) * HS + d] = acc[mt][nt][r] + bv;
            }
        }
    }
}

// ---------------------------------------------------------------------------
// Kernel 2: causal relu attention. grid(96, 4), block 256 (8 waves).
// Wave w handles the q-tile pair i0 = 2*(blockIdx.y*8 + w), i0+1 of head
// blockIdx.x. K and V fragments are shared across the pair (2x wmma/vmem).
// Tile 0 at j == i0+1 is fully masked by the causal predicate (adds zero).
// ---------------------------------------------------------------------------
__global__ __launch_bounds__(256, 2)
void attn_relu_causal_f32wmma(const float* __restrict__ qkv,
                              float* __restrict__ out) {
    __shared__ float Sld[8][512];            // per-wave 2x(16x16) S relayout

    const int lane = threadIdx.x & 31;
    const int wave = threadIdx.x >> 5;
    const int hl   = lane >> 4;
    const int l15  = lane & 15;

    const int bh = blockIdx.x;                       // 0..95  (b*12 + h)
    const int i0 = (blockIdx.y * 8 + wave) * 2;      // q tile pair base, 0..62

    const float* qp = qkv;
    const float* kp = qkv + (size_t)1 * BT * CIN;
    const float* vp = qkv + (size_t)2 * BT * CIN;
    const size_t base = (size_t)bh * TT * HS;

    // Q fragments for both 16x64 tiles: 2 x 16 K-chunks of 4
    v2f qf[2][16];
    {
        const float* qrow = qp + base + (size_t)(i0 * 16 + l15) * HS;
#pragma unroll
        for (int ti = 0; ti < 2; ++ti)
#pragma unroll
            for (int kc = 0; kc < 16; ++kc)
                qf[ti][kc] = *(const v2f*)(qrow + (size_t)ti * 16 * HS +
                                           kc * 4 + 2 * hl);
    }

    v8f y[2][4] = {};
    for (int j = 0; j <= i0 + 1; ++j) {
        // S_ti = Q_ti (16x64) . K_j^T (64x16); K fragments shared
        v8f s0 = {}, s1 = {};
        const float* krow = kp + base + (size_t)(j * 16 + l15) * HS;
#pragma unroll
        for (int kc = 0; kc < 16; ++kc) {
            v2f bf = *(const v2f*)(krow + kc * 4 + 2 * hl);
            s0 = wmma_f32(qf[0][kc], bf, s0);
            s1 = wmma_f32(qf[1][kc], bf, s1);
        }
        // scale + causal mask + relu, write C-layout into LDS
        const int n = j * 16 + l15;                  // key index
#pragma unroll
        for (int r = 0; r < 8; ++r) {
            const int m0_ = (i0 + 0) * 16 + r + 8 * hl;
            const int m1_ = (i0 + 1) * 16 + r + 8 * hl;
            float v0 = s0[r] * SCALE_;
            float v1 = s1[r] * SCALE_;
            v0 = (m0_ >= n) ? fmaxf(v0, 0.0f) : 0.0f;
            v1 = (m1_ >= n) ? fmaxf(v1, 0.0f) : 0.0f;
            Sld[wave][(r + 8 * hl) * 16 + l15]       = v0;
            Sld[wave][256 + (r + 8 * hl) * 16 + l15] = v1;
        }
        __builtin_amdgcn_wave_barrier();             // order LDS store -> load

        // Y_ti += S_ti (16x16) . V_j (16x64); V fragments shared
        const float* vbase = vp + base + (size_t)(j * 16) * HS;
#pragma unroll
        for (int kc2 = 0; kc2 < 4; ++kc2) {
            v2f sa0 = *(const v2f*)(&Sld[wave][l15 * 16 + kc2 * 4 + 2 * hl]);
            v2f sa1 = *(const v2f*)(&Sld[wave][256 + l15 * 16 + kc2 * 4 + 2 * hl]);
            const int row0 = kc2 * 4 + 2 * hl;
#pragma unroll
            for (int nt = 0; nt < 4; ++nt) {
                v2f vb;
                vb.x = vbase[(size_t)row0 * HS + nt * 16 + l15];
                vb.y = vbase[(size_t)(row0 + 1) * HS + nt * 16 + l15];
                y[0][nt] = wmma_f32(sa0, vb, y[0][nt]);
                y[1][nt] = wmma_f32(sa1, vb, y[1][nt]);
            }
        }
        __builtin_amdgcn_wave_barrier();             // keep S WAR ordered
    }

    // write out [B, T, nh*hs]
    const int b_ = bh / NH, h = bh % NH;
#pragma unroll
    for (int ti = 0; ti < 2; ++ti) {
#pragma unroll
        for (int nt = 0; nt < 4; ++nt) {
            const int d = nt * 16 + l15;
#pragma unroll
            for (int r = 0; r < 8; ++r) {
                const int t = (i0 + ti) * 16 + r + 8 * hl;
                out[(size_t)(b_ * TT + t) * CIN + h * HS + d] = y[ti][nt][r];
            }
        }
    }
}

extern "C" void kernel_launch(void* const* d_in, const int* in_sizes, int n_in,
                              void* d_out, int out_size, void* d_ws, size_t ws_size,
                              hipStream_t stream) {
    const float* x    = (const float*)d_in[0];   // [8,1024,768]
    const float* W    = (const float*)d_in[1];   // [768,2304]
    const float* bias = (const float*)d_in[2];   // [2304]
    float* out = (float*)d_out;                  // [8,1024,768]
    float* ws  = (float*)d_ws;                   // needs 3*8192*768*4 = 75.5 MB

    dim3 g1(BT / 512, C3 / 64);                  // (16, 36)
    attn_qkv_gemm_f32wmma<<<g1, 256, 0, stream>>>(x, W, bias, ws);

    dim3 g2(8 * NH, 4);                          // (96, 4)
    attn_relu_causal_f32wmma<<<g2, 256, 0, stream>>>(ws, out);
}